// ContrastiveLoss_82884278878350
// MI455X (gfx1250) — compile-verified
//
#include <hip/hip_runtime.h>
#include <hip/hip_bf16.h>
#include <math.h>

// N = 4096 rows, D = 1024 cols, fp32. Output: single fp32 scalar.
// Algorithm: avoid the 4096x4096 similarity matrix entirely; only its
// diagonal, row sums and column sums are needed (O(N*D), memory bound).

#define NROWS 4096
#define DCOLS 1024
#define EPSN  1e-18f

typedef float v2f __attribute__((ext_vector_type(2)));
typedef float v8f __attribute__((ext_vector_type(8)));

// ---------------------------------------------------------------------------
// fp32 WMMA helper: D = A(16x4) * B(4x16) + C, exact fp32 (V_WMMA_F32_16X16X4_F32)
// ---------------------------------------------------------------------------
__device__ __forceinline__ v8f wmma_f32_k4(v2f a, v2f b, v8f c) {
    return __builtin_amdgcn_wmma_f32_16x16x4_f32(
        /*neg_a=*/false, a, /*neg_b=*/false, b,
        /*c_mod=*/(short)0, c, /*reuse_a=*/false, /*reuse_b=*/false);
}

// Extract diagonal of the 16x16 fp32 accumulator via LDS.
// C layout (ISA 7.12.2): VGPR r, lanes 0-15 -> (M=r, N=lane); lanes 16-31 -> (M=r+8, N=lane-16)
__device__ __forceinline__ float extract_diag(const v8f& c, float* sC, int lane) {
    __syncthreads();
#pragma unroll
    for (int r = 0; r < 8; ++r) {
        int M = (lane < 16) ? r : r + 8;
        sC[M * 16 + (lane & 15)] = c[r];
    }
    __syncthreads();
    float d = 0.0f;
    if (lane < 16) d = sC[lane * 16 + lane];
    return d;
}

// ---------------------------------------------------------------------------
// Pass 1: per-16-row-tile Gram diagonals via WMMA.
//   dvv[kseg][r] += V_r . V_r   dva += V_r . A_r   daa += A_r . A_r  (k-range per seg)
// grid: (256 tiles, 4 k-segments), 32 threads (1 wave) per block.
// ---------------------------------------------------------------------------
#define SROW 68  // padded LDS row stride (floats): 4m % 64 distinct banks, 16B aligned

__global__ __launch_bounds__(32)
void diag_wmma_kernel(const float* __restrict__ V, const float* __restrict__ A,
                      float* __restrict__ dvv, float* __restrict__ dva,
                      float* __restrict__ daa) {
    __shared__ float sV[16 * SROW];
    __shared__ float sA[16 * SROW];
    __shared__ float sC[256];

    const int tile = blockIdx.x;            // 0..255
    const int kseg = blockIdx.y;            // 0..3
    const int r0   = tile * 16;
    const int lane = threadIdx.x;           // 0..31

    v8f cvv = {};
    v8f cva = {};
    v8f caa = {};

    const int m    = lane & 15;             // matrix row within tile
    const int koff = (lane >> 4) * 2;       // K offset 0 or 2 (ISA A/B frag layout)

    const int kbeg = kseg * 256;
    for (int k0 = kbeg; k0 < kbeg + 256; k0 += 64) {
        // cooperative load of 16x64 fp32 chunk of V and A into LDS (coalesced float4)
#pragma unroll
        for (int t = 0; t < 8; ++t) {
            int linear = t * 32 + lane;     // 0..255
            int row    = linear >> 4;       // 0..15
            int c4     = linear & 15;       // 0..15 (float4 index)
            size_t gofs = (size_t)(r0 + row) * DCOLS + (size_t)k0 + (size_t)c4 * 4;
            float4 v4 = *(const float4*)(V + gofs);
            float4 a4 = *(const float4*)(A + gofs);
            *(float4*)(sV + row * SROW + c4 * 4) = v4;
            *(float4*)(sA + row * SROW + c4 * 4) = a4;
        }
        __syncthreads();

        // 16 K-steps of 4; A-frag and B-frag carry identical per-lane values
        // (A 16x4: lane m holds K=koff,koff+1; B 4x16: lane n=m holds same Ks)
#pragma unroll
        for (int kk = 0; kk < 64; kk += 4) {
            v2f fv, fa;
            fv.x = sV[m * SROW + kk + koff];
            fv.y = sV[m * SROW + kk + koff + 1];
            fa.x = sA[m * SROW + kk + koff];
            fa.y = sA[m * SROW + kk + koff + 1];
            cvv = wmma_f32_k4(fv, fv, cvv);
            cva = wmma_f32_k4(fv, fa, cva);
            caa = wmma_f32_k4(fa, fa, caa);
        }
        __syncthreads();
    }

    float d_vv = extract_diag(cvv, sC, lane);
    float d_va = extract_diag(cva, sC, lane);
    float d_aa = extract_diag(caa, sC, lane);

    if (lane < 16) {
        int r = r0 + lane;
        dvv[kseg * NROWS + r] = d_vv;
        dva[kseg * NROWS + r] = d_va;
        daa[kseg * NROWS + r] = d_aa;
    }
}

// ---------------------------------------------------------------------------
// Pass 2: combine k-segment partials -> rv = 1/sqrt(|V|^2+eps), ra, pos (diag of S)
// ---------------------------------------------------------------------------
__global__ __launch_bounds__(256)
void norms_kernel(const float* __restrict__ dvv, const float* __restrict__ dva,
                  const float* __restrict__ daa,
                  float* __restrict__ rv, float* __restrict__ ra, float* __restrict__ pos) {
    int r = blockIdx.x * 256 + threadIdx.x;   // 0..4095
    float svv = 0.f, sva = 0.f, saa = 0.f;
#pragma unroll
    for (int s = 0; s < 4; ++s) {
        svv += dvv[s * NROWS + r];
        sva += dva[s * NROWS + r];
        saa += daa[s * NROWS + r];
    }
    float irv = 1.0f / sqrtf(svv + EPSN);
    float ira = 1.0f / sqrtf(saa + EPSN);
    rv[r] = irv;
    ra[r] = ira;
    pos[r] = sva * irv * ira;
}

// ---------------------------------------------------------------------------
// Pass 3: partial column sums of normalized rows.
// grid: (4 col-blocks of 256 cols, 64 row-chunks of 64 rows), 256 threads.
// ---------------------------------------------------------------------------
__global__ __launch_bounds__(256)
void colsum_partial_kernel(const float* __restrict__ V, const float* __restrict__ A,
                           const float* __restrict__ rv, const float* __restrict__ ra,
                           float* __restrict__ partV, float* __restrict__ partA) {
    int col   = blockIdx.x * 256 + threadIdx.x;   // 0..1023
    int chunk = blockIdx.y;                        // 0..63
    int rbeg  = chunk * 64;
    float aV = 0.f, aA = 0.f;
#pragma unroll 4
    for (int i = 0; i < 64; ++i) {
        int r = rbeg + i;
        aV += V[(size_t)r * DCOLS + col] * rv[r];
        aA += A[(size_t)r * DCOLS + col] * ra[r];
    }
    partV[chunk * DCOLS + col] = aV;
    partA[chunk * DCOLS + col] = aA;
}

// ---------------------------------------------------------------------------
// Pass 4: finalize column sums (fixed order => deterministic)
// ---------------------------------------------------------------------------
__global__ __launch_bounds__(256)
void colsum_final_kernel(const float* __restrict__ partV, const float* __restrict__ partA,
                         float* __restrict__ Vsum, float* __restrict__ Asum) {
    int col = blockIdx.x * 256 + threadIdx.x;     // 0..1023
    float sv = 0.f, sa = 0.f;
    for (int ch = 0; ch < 64; ++ch) {
        sv += partV[ch * DCOLS + col];
        sa += partA[ch * DCOLS + col];
    }
    Vsum[col] = sv;   // sum of normalized V rows
    Asum[col] = sa;   // sum of normalized A rows
}

// ---------------------------------------------------------------------------
// Pass 5: per-row loss terms.
//   rowsum_i = (V_i . Asum) * rv_i ; colsum_i = (A_i . Vsum) * ra_i
//   term_i = log(pos/rowsum') + log(pos/colsum')  with the i==0 special case.
// ---------------------------------------------------------------------------
__global__ __launch_bounds__(256)
void rowterm_kernel(const float* __restrict__ V, const float* __restrict__ A,
                    const float* __restrict__ rv, const float* __restrict__ ra,
                    const float* __restrict__ pos,
                    const float* __restrict__ Vsum, const float* __restrict__ Asum,
                    float* __restrict__ term) {
    __shared__ float s1[256];
    __shared__ float s2[256];
    int row = blockIdx.x;
    int t   = threadIdx.x;

    float4 v4  = ((const float4*)(V + (size_t)row * DCOLS))[t];
    float4 a4  = ((const float4*)(A + (size_t)row * DCOLS))[t];
    float4 as4 = ((const float4*)Asum)[t];
    float4 vs4 = ((const float4*)Vsum)[t];

    s1[t] = v4.x * as4.x + v4.y * as4.y + v4.z * as4.z + v4.w * as4.w;
    s2[t] = a4.x * vs4.x + a4.y * vs4.y + a4.z * vs4.z + a4.w * vs4.w;
    __syncthreads();
    for (int off = 128; off > 0; off >>= 1) {
        if (t < off) { s1[t] += s1[t + off]; s2[t] += s2[t + off]; }
        __syncthreads();
    }
    if (t == 0) {
        float rowsum = s1[0] * rv[row];
        float colsum = s2[0] * ra[row];
        float p = pos[row];
        // i != 0: pos + (rowsum - pos) = rowsum ; i == 0: pos + rowsum
        float tV = (row == 0) ? logf(p / (p + rowsum)) : logf(p / rowsum);
        float tA = (row == 0) ? logf(p / (p + colsum)) : logf(p / colsum);
        term[row] = tV + tA;
    }
}

// ---------------------------------------------------------------------------
// Pass 6: deterministic final reduction of 4096 terms -> scalar
// ---------------------------------------------------------------------------
__global__ __launch_bounds__(256)
void final_kernel(const float* __restrict__ term, float* __restrict__ out) {
    __shared__ float s[256];
    int t = threadIdx.x;
    float acc = 0.f;
    for (int i = t; i < NROWS; i += 256) acc += term[i];
    s[t] = acc;
    __syncthreads();
    for (int off = 128; off > 0; off >>= 1) {
        if (t < off) s[t] += s[t + off];
        __syncthreads();
    }
    if (t == 0) out[0] = (-1.0f / (float)NROWS) * s[0];
}

// ---------------------------------------------------------------------------
extern "C" void kernel_launch(void* const* d_in, const int* in_sizes, int n_in,
                              void* d_out, int out_size, void* d_ws, size_t ws_size,
                              hipStream_t stream) {
    (void)in_sizes; (void)n_in; (void)out_size; (void)ws_size;
    const float* V = (const float*)d_in[0];   // back_VF [4096,1024] fp32
    const float* A = (const float*)d_in[1];   // back_AF [4096,1024] fp32

    float* ws   = (float*)d_ws;               // ~780 KB used
    float* rv   = ws;                         // 4096
    float* ra   = rv + NROWS;                 // 4096
    float* pos  = ra + NROWS;                 // 4096
    float* term = pos + NROWS;                // 4096
    float* Vsum = term + NROWS;               // 1024
    float* Asum = Vsum + DCOLS;               // 1024
    float* dvv  = Asum + DCOLS;               // 4*4096
    float* dva  = dvv + 4 * NROWS;            // 4*4096
    float* daa  = dva + 4 * NROWS;            // 4*4096
    float* partV = daa + 4 * NROWS;           // 64*1024
    float* partA = partV + 64 * DCOLS;        // 64*1024

    diag_wmma_kernel<<<dim3(NROWS / 16, 4), 32, 0, stream>>>(V, A, dvv, dva, daa);
    norms_kernel<<<NROWS / 256, 256, 0, stream>>>(dvv, dva, daa, rv, ra, pos);
    colsum_partial_kernel<<<dim3(DCOLS / 256, 64), 256, 0, stream>>>(V, A, rv, ra, partV, partA);
    colsum_final_kernel<<<DCOLS / 256, 256, 0, stream>>>(partV, partA, Vsum, Asum);
    rowterm_kernel<<<NROWS, 256, 0, stream>>>(V, A, rv, ra, pos, Vsum, Asum, term);
    final_kernel<<<1, 256, 0, stream>>>(term, (float*)d_out);
}